// PCAGarmentEncoderSeg_86466281603624
// MI455X (gfx1250) — compile-verified
//
#include <hip/hip_runtime.h>
#include <hip/hip_bf16.h>

// ---- fixed problem shapes (from the reference) ----
#define NB_    8
#define T_     8
#define NPTS   6890
#define C_     128
#define KCLS   7
#define DSUM   512
#define PCAD   64
#define VV     7702
#define KC     (VV * 3)        // 23106
#define NSEL   1722            // N // 4
#define BT     (NB_ * T_)      // 64
#define EPS_   1e-5f

typedef __attribute__((ext_vector_type(2))) float v2f;
typedef __attribute__((ext_vector_type(8))) float v8f;

// ---------------------------------------------------------------------------
// Kernel 1: per-batch stable segmentation scan.
// One 256-thread block per batch b. Each thread owns a contiguous 27-point
// segment (stability = index order preserved), computes argmax labels, packs
// its hit mask in a 32-bit register, block-wide exclusive scan in LDS, then
// writes selected indices to ws.
// ---------------------------------------------------------------------------
__global__ __launch_bounds__(256) void seg_scan_kernel(
    const float* __restrict__ sem_logits, const int* __restrict__ glabel_p,
    int* __restrict__ sel, int* __restrict__ cnt_out)
{
  const int b = blockIdx.x;
  const int t = threadIdx.x;
  const int glabel = glabel_p[0];
  const int SEG = 27;                       // ceil(6890/256)
  const int start = t * SEG;
  const int end = min(start + SEG, NPTS);
  const float* L = sem_logits + (size_t)b * NPTS * KCLS;

  unsigned mbits = 0;
  int cnt = 0;
  for (int i = start; i < end; ++i) {
    const float* p = L + (size_t)i * KCLS;
    float best = p[0];
    int lab = 0;
#pragma unroll
    for (int k = 1; k < KCLS; ++k) {        // strict '>' == first-max (jnp.argmax)
      float v = p[k];
      if (v > best) { best = v; lab = k; }
    }
    if (lab == glabel) { mbits |= (1u << (i - start)); ++cnt; }
  }

  __shared__ int sscan[256];
  sscan[t] = cnt;
  __syncthreads();
  for (int off = 1; off < 256; off <<= 1) {
    int v = (t >= off) ? sscan[t - off] : 0;
    __syncthreads();
    sscan[t] += v;
    __syncthreads();
  }
  const int total = sscan[255];
  int pos = sscan[t] - cnt;                 // exclusive prefix
  for (int i = start; i < end; ++i) {
    if (mbits & (1u << (i - start))) {
      if (pos < NSEL) sel[b * NSEL + pos] = i;
      ++pos;
    }
  }
  if (t == 0) cnt_out[b] = min(total, NSEL);
}

// ---------------------------------------------------------------------------
// Kernel 2: gather x (3ch) and feature (128ch) at selected indices, zero-pad.
// One 128-thread block per (b, j); thread c handles channel c.
// ---------------------------------------------------------------------------
__global__ __launch_bounds__(128) void seg_gather_kernel(
    const float* __restrict__ x, const float* __restrict__ feature,
    const int* __restrict__ sel, const int* __restrict__ cnt,
    float* __restrict__ gv, float* __restrict__ gf)
{
  const int bj = blockIdx.x;
  const int b = bj / NSEL;
  const int j = bj - b * NSEL;
  const int c = threadIdx.x;
  const bool valid = j < cnt[b];
  const int idx = valid ? sel[bj] : 0;

  float fv = valid ? feature[((size_t)b * C_ + c) * NPTS + idx] : 0.0f;
  gf[(size_t)bj * C_ + c] = fv;
  if (c < 3) {
    float xv = valid ? x[((size_t)b * NPTS + idx) * 3 + c] : 0.0f;
    gv[(size_t)bj * 3 + c] = xv;
  }
}

// ---------------------------------------------------------------------------
// Kernel 3: max-over-T + 3-layer MLP (BN eval + ReLU). One 512-thread block.
// ---------------------------------------------------------------------------
__global__ __launch_bounds__(512) void mlp_kernel(
    const float* __restrict__ gsum,
    const float* __restrict__ W1, const float* __restrict__ b1,
    const float* __restrict__ g1, const float* __restrict__ be1,
    const float* __restrict__ rm1, const float* __restrict__ rv1,
    const float* __restrict__ W2, const float* __restrict__ b2,
    const float* __restrict__ g2, const float* __restrict__ be2,
    const float* __restrict__ rm2, const float* __restrict__ rv2,
    const float* __restrict__ W3, const float* __restrict__ b3,
    float* __restrict__ coeff)
{
  __shared__ float gs[NB_ * DSUM];   // 16 KB
  __shared__ float h1[NB_ * 128];    // 4 KB
  __shared__ float h2[NB_ * 64];     // 2 KB
  const int t = threadIdx.x;

  // gs = max over T
  for (int b = 0; b < NB_; ++b) {
    float m = -3.402823466e38f;
    for (int tt = 0; tt < T_; ++tt)
      m = fmaxf(m, gsum[((size_t)b * T_ + tt) * DSUM + t]);
    gs[b * DSUM + t] = m;
  }
  __syncthreads();

  // layer 1: 8x128 outputs (2 per thread)
  for (int e = t; e < NB_ * 128; e += 512) {
    int b = e >> 7, o = e & 127;
    float acc = b1[o];
    const float* w = W1 + (size_t)o * DSUM;
    const float* g = gs + b * DSUM;
    for (int k = 0; k < DSUM; ++k) acc += g[k] * w[k];
    acc = (acc - rm1[o]) * (1.0f / sqrtf(rv1[o] + EPS_)) * g1[o] + be1[o];
    h1[b * 128 + o] = fmaxf(acc, 0.0f);
  }
  __syncthreads();

  // layer 2: 8x64 = 512 outputs (1 per thread)
  {
    int b = t >> 6, o = t & 63;
    float acc = b2[o];
    const float* w = W2 + (size_t)o * 128;
    const float* hh = h1 + b * 128;
    for (int k = 0; k < 128; ++k) acc += hh[k] * w[k];
    acc = (acc - rm2[o]) * (1.0f / sqrtf(rv2[o] + EPS_)) * g2[o] + be2[o];
    h2[b * 64 + o] = fmaxf(acc, 0.0f);
  }
  __syncthreads();

  // layer 3: coeff = h2 @ W3^T + b3
  {
    int b = t >> 6, o = t & 63;
    float acc = b3[o];
    const float* w = W3 + (size_t)o * 64;
    const float* hh = h2 + b * 64;
    for (int k = 0; k < 64; ++k) acc += hh[k] * w[k];
    coeff[b * PCAD + o] = acc;
  }
}

// ---------------------------------------------------------------------------
// Kernel 4: tpose = (coeff[8x64] @ pca_comp[64x23106] + mean) * scale
// via V_WMMA_F32_16X16X4_F32. One 16x16 output tile per wave; M padded 8->16
// with zero A rows; K=64 -> 16 WMMA steps. EXEC stays all-1 through the WMMA
// loop (OOB columns clamp-loaded, predicated only at the store).
// ---------------------------------------------------------------------------
__global__ __launch_bounds__(128) void tpose_wmma_kernel(
    const float* __restrict__ coeff, const float* __restrict__ pca_comp,
    const float* __restrict__ pca_mean, const float* __restrict__ pca_scale,
    float* __restrict__ tpose)
{
  const int lane = threadIdx.x & 31;
  const int wave = threadIdx.x >> 5;
  const int ntiles = (KC + 15) / 16;         // 1445
  const int tile = blockIdx.x * 4 + wave;    // wave-uniform
  if (tile >= ntiles) return;

  const int nl = lane & 15;
  const int khalf = (lane >> 4) << 1;        // 0 or 2: K pair handled by this half-wave

  // A fragments: lane holds coeff[m][kk], coeff[m][kk+1]; rows >= 8 are zero pad.
  v2f afrag[16];
#pragma unroll
  for (int kb = 0; kb < 16; ++kb) {
    const int kk = kb * 4 + khalf;
    float a0 = (nl < NB_) ? coeff[nl * PCAD + kk]     : 0.0f;
    float a1 = (nl < NB_) ? coeff[nl * PCAD + kk + 1] : 0.0f;
    v2f a; a.x = a0; a.y = a1;
    afrag[kb] = a;
  }

  const int ncol = tile * 16 + nl;
  const int ncs = (ncol < KC) ? ncol : (KC - 1);   // clamp: GEMM columns independent

  v8f acc = {};
#pragma unroll
  for (int kb = 0; kb < 16; ++kb) {
    const int kk = kb * 4 + khalf;
    v2f bfrag;
    bfrag.x = pca_comp[(size_t)kk * KC + ncs];
    bfrag.y = pca_comp[(size_t)(kk + 1) * KC + ncs];
    acc = __builtin_amdgcn_wmma_f32_16x16x4_f32(
        /*neg_a=*/false, afrag[kb], /*neg_b=*/false, bfrag,
        /*c_mod=*/(short)0, acc, /*reuse_a=*/false, /*reuse_b=*/false);
  }

  // D layout: VGPR r -> M=r on lanes 0-15 (M=r+8 on lanes 16-31 = zero pad rows).
  if (ncol < KC && lane < 16) {
    const float mn = pca_mean[ncol];
    const float sc = pca_scale[ncol];
#pragma unroll
    for (int r = 0; r < 8; ++r)
      tpose[(size_t)r * KC + ncol] = (acc[r] + mn) * sc;
  }
}

// ---------------------------------------------------------------------------
extern "C" void kernel_launch(void* const* d_in, const int* in_sizes, int n_in,
                              void* d_out, int out_size, void* d_ws, size_t ws_size,
                              hipStream_t stream) {
  const float* x        = (const float*)d_in[0];
  const float* sem      = (const float*)d_in[1];
  const float* feature  = (const float*)d_in[2];
  const float* gsum     = (const float*)d_in[3];
  const float* W1  = (const float*)d_in[4];
  const float* b1  = (const float*)d_in[5];
  const float* g1  = (const float*)d_in[6];
  const float* be1 = (const float*)d_in[7];
  const float* rm1 = (const float*)d_in[8];
  const float* rv1 = (const float*)d_in[9];
  const float* W2  = (const float*)d_in[10];
  const float* b2  = (const float*)d_in[11];
  const float* g2  = (const float*)d_in[12];
  const float* be2 = (const float*)d_in[13];
  const float* rm2 = (const float*)d_in[14];
  const float* rv2 = (const float*)d_in[15];
  const float* W3  = (const float*)d_in[16];
  const float* b3  = (const float*)d_in[17];
  const float* pca_comp  = (const float*)d_in[18];
  const float* pca_mean  = (const float*)d_in[19];
  const float* pca_scale = (const float*)d_in[20];
  const int*   glabel    = (const int*)d_in[21];
  // d_in[22] = n (==1722), shape constant; grid dims use compile-time NSEL.

  // output layout: gv | gf | coeff | tpose (flat, return order)
  float* out   = (float*)d_out;
  float* gv    = out;                                   // 64*1722*3
  float* gf    = gv + (size_t)BT * NSEL * 3;            // 64*1722*128
  float* coeff = gf + (size_t)BT * NSEL * C_;           // 8*64
  float* tpose = coeff + (size_t)NB_ * PCAD;            // 8*23106

  // workspace: selected indices + counts
  int* sel = (int*)d_ws;                                // BT*NSEL ints
  int* cnt = sel + (size_t)BT * NSEL;                   // BT ints

  seg_scan_kernel<<<BT, 256, 0, stream>>>(sem, glabel, sel, cnt);
  seg_gather_kernel<<<BT * NSEL, 128, 0, stream>>>(x, feature, sel, cnt, gv, gf);
  mlp_kernel<<<1, 512, 0, stream>>>(gsum, W1, b1, g1, be1, rm1, rv1,
                                    W2, b2, g2, be2, rm2, rv2, W3, b3, coeff);
  const int ntiles = (KC + 15) / 16;                    // 1445
  tpose_wmma_kernel<<<(ntiles + 3) / 4, 128, 0, stream>>>(coeff, pca_comp,
                                                          pca_mean, pca_scale, tpose);
}